// WrapperModel_53240414601491
// MI455X (gfx1250) — compile-verified
//
#include <hip/hip_runtime.h>
#include <hip/hip_bf16.h>
#include <limits.h>

// ---------------------------------------------------------------------------
// MI455X / gfx1250 implementation.
//  - Stage 1 (bandwidth roofline): single fused pass over the 8192^2 image
//    producing row/col "has variation" flags (256 MB read ~= 11us @ 23.3TB/s),
//    with global_prefetch_b8 ahead of the stream.
//  - Stage 2: tiny bounds / bilinear-resize / dilation kernels.
//  - Stage 3: all 5 conv layers as implicit GEMM on V_WMMA_F32_16X16X4_F32,
//    weights staged to LDS (transposed [k][n]) via GLOBAL_LOAD_ASYNC_TO_LDS
//    (ASYNCcnt / s_wait_asynccnt), fused bias+BN+ReLU epilogue.
//  - Stage 4: global-average-pool + 2-way head + argmax.
// ---------------------------------------------------------------------------

typedef float v2f __attribute__((ext_vector_type(2)));
typedef float v8f __attribute__((ext_vector_type(8)));

#define IMG_W 8192
#define IMG_H 8192
#define OUTR  256
#define KC    144   // K-chunk for LDS weight staging: 144*64*4B = 36 KB

typedef __attribute__((address_space(1))) int* gptr_i32;
typedef __attribute__((address_space(3))) int* lptr_i32;

// ---- CDNA5 async global->LDS copy (guarded; falls back to plain copy) -----
__device__ inline void async_g2l_b32(const float* g, float* l) {
#if __has_builtin(__builtin_amdgcn_global_load_async_to_lds_b32)
    __builtin_amdgcn_global_load_async_to_lds_b32(
        (gptr_i32)g, (lptr_i32)l, /*offset=*/0, /*cpol=*/0);
#else
    *l = *g;
#endif
}
__device__ inline void async_g2l_wait() {
#if __has_builtin(__builtin_amdgcn_s_wait_asynccnt)
    __builtin_amdgcn_s_wait_asynccnt(0);
#endif
}

// ---------------- flags zeroing -------------------------------------------
__global__ void k_zero_u32(unsigned* p, int n) {
    int i = blockIdx.x * blockDim.x + threadIdx.x;
    if (i < n) p[i] = 0u;
}

// ---------------- fused row/col variation flags ----------------------------
// col_var[c] = OR_r (img[r][c] != img[0][c]);  row_var[r] = OR_c (img[r][c] != img[r][0])
__global__ void k_varflags(const float* __restrict__ img,
                           unsigned* __restrict__ colv,
                           unsigned* __restrict__ rowv) {
    int c  = blockIdx.x * 256 + threadIdx.x;
    int r0 = blockIdx.y * 256;
    float cref = img[c];                       // img[0][c]
    unsigned colacc = 0u;
    for (int rr = 0; rr < 256; ++rr) {
        int r = r0 + rr;
        long base = (long)r * IMG_W;
        // stream-ahead prefetch (global_prefetch_b8); speculative, OOB-safe
        __builtin_prefetch(img + base + (long)8 * IMG_W + c, 0, 1);
        float v    = img[base + c];
        float rref = img[base];                // img[r][0] (broadcast load)
        colacc |= (v != cref) ? 1u : 0u;
        int any = __any(v != rref);
        if (((threadIdx.x & 31) == 0) && any) atomicOr(&rowv[r], 1u);
    }
    if (colacc) atomicOr(&colv[c], 1u);
}

// ---------------- crop bounds ---------------------------------------------
__global__ void k_bounds(const unsigned* __restrict__ colv,
                         const unsigned* __restrict__ rowv,
                         int* __restrict__ bounds) {
    __shared__ int s[4];   // [minRow, maxRow, minCol, maxCol]
    if (threadIdx.x == 0) { s[0] = INT_MAX; s[1] = -1; s[2] = INT_MAX; s[3] = -1; }
    __syncthreads();
    for (int i = threadIdx.x; i < IMG_W; i += blockDim.x) {
        if (rowv[i]) { atomicMin(&s[0], i); atomicMax(&s[1], i); }
        if (colv[i]) { atomicMin(&s[2], i); atomicMax(&s[3], i); }
    }
    __syncthreads();
    if (threadIdx.x == 0) {
        bounds[0] = (s[1] < 0) ? 0          : s[0];  // top
        bounds[1] = (s[1] < 0) ? IMG_H - 1  : s[1];  // bottom
        bounds[2] = (s[3] < 0) ? 0          : s[2];  // left
        bounds[3] = (s[3] < 0) ? IMG_W - 1  : s[3];  // right
    }
}

// ---------------- crop + bilinear resize, store im = 255 - resized ---------
__device__ inline void axis_coord(int lo, int hi, int j, int& i0, int& i1, float& w) {
    float size = (float)(hi - lo + 1);
    float c  = (float)lo + ((float)j + 0.5f) * size / (float)OUTR - 0.5f;
    float c0 = floorf(c);
    w = c - c0;
    int ic = (int)c0;
    i0 = min(max(ic,     lo), hi);
    i1 = min(max(ic + 1, lo), hi);
}

__global__ void k_resize(const float* __restrict__ img,
                         const int* __restrict__ bounds,
                         float* __restrict__ im) {
    int idx = blockIdx.x * blockDim.x + threadIdx.x;     // 65536 threads
    int yo = idx >> 8, xo = idx & 255;
    int top = bounds[0], bottom = bounds[1], left = bounds[2], right = bounds[3];
    int y0, y1, x0, x1; float wy, wx;
    axis_coord(top,  bottom, yo, y0, y1, wy);
    axis_coord(left, right,  xo, x0, x1, wx);
    float a = img[(long)y0 * IMG_W + x0];
    float b = img[(long)y0 * IMG_W + x1];
    float c = img[(long)y1 * IMG_W + x0];
    float d = img[(long)y1 * IMG_W + x1];
    float tr = a * (1.0f - wx) + b * wx;
    float br = c * (1.0f - wx) + d * wx;
    float v  = tr * (1.0f - wy) + br * wy;
    im[idx] = 255.0f - v;
}

// ---------------- 3x3 rate-2 dilation (SAME, ones kernel) + invert+clip ----
__global__ void k_dilate(const float* __restrict__ im, float* __restrict__ xp) {
    int idx = blockIdx.x * blockDim.x + threadIdx.x;
    int y = idx >> 8, x = idx & 255;
    float s = 0.0f;
    #pragma unroll
    for (int ty = 0; ty < 3; ++ty) {
        int iy = y + (ty - 1) * 2;
        if (iy < 0 || iy >= OUTR) continue;
        #pragma unroll
        for (int tx = 0; tx < 3; ++tx) {
            int ix = x + (tx - 1) * 2;
            if (ix < 0 || ix >= OUTR) continue;
            s += im[iy * OUTR + ix];
        }
    }
    s = fminf(fmaxf(s, 0.0f), 255.0f);
    xp[idx] = 255.0f - s;
}

// ---------------- implicit-GEMM conv block on V_WMMA_F32_16X16X4_F32 -------
// Conv2d(k=3, stride=2, pad=1) -> +bias -> BN(eval) -> ReLU.
// One wave per 16(M=pixels) x 16(N=channels) tile; blockDim=128 (4 waves = 64 ch).
// Weights staged transposed into LDS as sB[k][n] via async global->LDS copies
// so the B-operand fetch is a contiguous ds_load shared by all 4 waves.
// A (16x4, MxK): lane L -> m = L&15, k-pair base = 2*(L>>4);  B (4x16) symmetric.
// C/D (16x16): lane L, vgpr r -> m = 8*(L>>4)+r, n = L&15.
__device__ inline float fetchA(const float* __restrict__ in, int k, int K,
                               int oy, int ox, int IH, int IW) {
    float av = 0.0f;
    if (k < K) {
        int c  = k / 9;
        int kk = k - c * 9;
        int ky = kk / 3;
        int kx = kk - ky * 3;
        int iy = oy * 2 - 1 + ky;
        int ix = ox * 2 - 1 + kx;
        if ((unsigned)iy < (unsigned)IH && (unsigned)ix < (unsigned)IW)
            av = in[((long)c * IH + iy) * IW + ix];
    }
    return av;
}

__global__ void k_conv_wmma(const float* __restrict__ in,  const float* __restrict__ w,
                            const float* __restrict__ bias,
                            const float* __restrict__ gamma, const float* __restrict__ beta,
                            const float* __restrict__ mean,  const float* __restrict__ var,
                            float* __restrict__ out,
                            int Cin, int IH, int IW, int OH, int OW) {
    __shared__ float sB[KC * 64];          // [kRel][n], 36 KB

    const int K    = Cin * 9;
    const int Kpad = (K + 3) & ~3;
    const int HW   = OH * OW;

    int tid   = threadIdx.x;
    int lane  = tid & 31;
    int mtile = blockIdx.x;

    int m0    = mtile * 16;
    int mlane = lane & 15;                 // A-fragment row
    int khalf = (lane >> 4) * 2;           // A/B k offset within 4-wide step
    int n     = (tid >> 5) * 16 + (lane & 15);  // output channel, 0..63

    int pix = m0 + mlane;
    int oy  = pix / OW;
    int ox  = pix - oy * OW;

    v8f acc = {};
    for (int kc0 = 0; kc0 < Kpad; kc0 += KC) {
        int kl = min(KC, Kpad - kc0);      // chunk length (multiple of 4)

        // ---- stage B chunk to LDS: sB[kRel*64 + n] = w[n*K + (kc0+kRel)] ----
        __syncthreads();                   // protect previous chunk's readers
        for (int e = tid; e < kl * 64; e += 128) {
            int kRel = e >> 6;
            int nn   = e & 63;
            int k    = kc0 + kRel;
            if (k < K) async_g2l_b32(&w[(long)nn * K + k], &sB[e]);
            else       sB[e] = 0.0f;       // K padding
        }
        async_g2l_wait();                  // s_wait_asynccnt 0
        __syncthreads();

        // ---- GEMM over the chunk --------------------------------------
        #pragma unroll 4
        for (int kr = 0; kr < kl; kr += 4) {
            int k0 = kc0 + kr + khalf;
            float a0 = fetchA(in, k0,     K, oy, ox, IH, IW);
            float a1 = fetchA(in, k0 + 1, K, oy, ox, IH, IW);
            v2f av; av.x = a0; av.y = a1;
            v2f bv; bv.x = sB[(kr + khalf) * 64 + n];
                    bv.y = sB[(kr + khalf + 1) * 64 + n];
            acc = __builtin_amdgcn_wmma_f32_16x16x4_f32(
                /*neg_a=*/false, av, /*neg_b=*/false, bv,
                /*c_mod=*/(short)0, acc, /*reuse_a=*/false, /*reuse_b=*/false);
        }
    }

    // fused epilogue: channel n is lane-constant across all 8 accum VGPRs
    float sc = gamma[n] * rsqrtf(var[n] + 1e-5f);
    float bi = bias[n];
    float mu = mean[n];
    float be = beta[n];
    int mrow0 = (lane >> 4) * 8;
    #pragma unroll
    for (int r = 0; r < 8; ++r) {
        int m = m0 + mrow0 + r;
        float v = acc[r] + bi;
        v = (v - mu) * sc + be;
        v = fmaxf(v, 0.0f);
        out[(long)n * HW + m] = v;
    }
}

// ---------------- global average pool + linear head + argmax ---------------
__global__ void k_head(const float* __restrict__ act,   // [64][8][8]
                       const float* __restrict__ hw,    // [2][64]
                       const float* __restrict__ hb,    // [2]
                       float* __restrict__ outp) {
    __shared__ float feat[64];
    int t = threadIdx.x;                   // 64 threads
    float s = 0.0f;
    for (int i = 0; i < 64; ++i) s += act[t * 64 + i];
    feat[t] = s * (1.0f / 64.0f);
    __syncthreads();
    if (t == 0) {
        float l0 = hb[0], l1 = hb[1];
        for (int c = 0; c < 64; ++c) {
            l0 += hw[c]      * feat[c];
            l1 += hw[64 + c] * feat[c];
        }
        outp[0] = (l1 > l0) ? 1.0f : 0.0f;   // jnp.argmax: first max wins
    }
}

// ---------------------------------------------------------------------------
extern "C" void kernel_launch(void* const* d_in, const int* in_sizes, int n_in,
                              void* d_out, int out_size, void* d_ws, size_t ws_size,
                              hipStream_t stream) {
    const float* image   = (const float*)d_in[0];
    const float* conv0_w = (const float*)d_in[1];
    const float* conv0_b = (const float*)d_in[2];
    const float* convs_w = (const float*)d_in[3];   // [4][64][64][3][3]
    const float* convs_b = (const float*)d_in[4];   // [4][64]
    const float* bn_g    = (const float*)d_in[5];   // [5][64]
    const float* bn_b    = (const float*)d_in[6];
    const float* bn_m    = (const float*)d_in[7];
    const float* bn_v    = (const float*)d_in[8];
    const float* head_w  = (const float*)d_in[9];   // [2][64]
    const float* head_b  = (const float*)d_in[10];  // [2]

    // workspace layout
    unsigned* colv   = (unsigned*)d_ws;             // 8192
    unsigned* rowv   = colv + IMG_W;                // 8192
    int*      bounds = (int*)(rowv + IMG_H);        // 4 (+pad)
    float*    im     = (float*)(bounds + 16);       // 256*256
    float*    xp     = im + OUTR * OUTR;            // 256*256
    float*    actA   = xp + OUTR * OUTR;            // up to 64*128*128
    float*    actB   = actA + 64 * 128 * 128;       // up to 64*64*64

    // 1) flags
    k_zero_u32<<<(2 * IMG_W + 255) / 256, 256, 0, stream>>>(colv, 2 * IMG_W);
    dim3 vg(IMG_W / 256, IMG_H / 256);
    k_varflags<<<vg, 256, 0, stream>>>(image, colv, rowv);
    k_bounds<<<1, 1024, 0, stream>>>(colv, rowv, bounds);

    // 2) crop-resize + dilation preprocessing
    k_resize<<<OUTR * OUTR / 256, 256, 0, stream>>>(image, bounds, im);
    k_dilate<<<OUTR * OUTR / 256, 256, 0, stream>>>(im, xp);

    // 3) conv stack (WMMA implicit GEMM); layer i uses bn_*[i*64 ...]
    // conv0: [1,256,256] -> [64,128,128]
    k_conv_wmma<<<(128 * 128) / 16, 128, 0, stream>>>(
        xp, conv0_w, conv0_b, bn_g + 0, bn_b + 0, bn_m + 0, bn_v + 0,
        actA, 1, 256, 256, 128, 128);
    // conv1: [64,128,128] -> [64,64,64]
    k_conv_wmma<<<(64 * 64) / 16, 128, 0, stream>>>(
        actA, convs_w + 0L * 64 * 64 * 9, convs_b + 0 * 64,
        bn_g + 64, bn_b + 64, bn_m + 64, bn_v + 64,
        actB, 64, 128, 128, 64, 64);
    // conv2: -> [64,32,32]
    k_conv_wmma<<<(32 * 32) / 16, 128, 0, stream>>>(
        actB, convs_w + 1L * 64 * 64 * 9, convs_b + 1 * 64,
        bn_g + 128, bn_b + 128, bn_m + 128, bn_v + 128,
        actA, 64, 64, 64, 32, 32);
    // conv3: -> [64,16,16]
    k_conv_wmma<<<(16 * 16) / 16, 128, 0, stream>>>(
        actA, convs_w + 2L * 64 * 64 * 9, convs_b + 2 * 64,
        bn_g + 192, bn_b + 192, bn_m + 192, bn_v + 192,
        actB, 64, 32, 32, 16, 16);
    // conv4: -> [64,8,8]
    k_conv_wmma<<<(8 * 8) / 16, 128, 0, stream>>>(
        actB, convs_w + 3L * 64 * 64 * 9, convs_b + 3 * 64,
        bn_g + 256, bn_b + 256, bn_m + 256, bn_v + 256,
        actA, 64, 16, 16, 8, 8);

    // 4) pool + head + argmax
    k_head<<<1, 64, 0, stream>>>(actA, head_w, head_b, (float*)d_out);
}